// DiscriminativeLoss_32229434589496
// MI455X (gfx1250) — compile-verified
//
#include <hip/hip_runtime.h>
#include <hip/hip_bf16.h>

typedef float v2f __attribute__((ext_vector_type(2)));
typedef float v8f __attribute__((ext_vector_type(8)));

namespace {
constexpr int KCLS   = 19;
constexpr int CHN    = 64;
constexpr int HW     = 512 * 512;        // 262144 = 1<<18
constexpr int NBATCH = 4;
constexpr int NPIX   = NBATCH * HW;      // 1048576
constexpr int PLANES = NBATCH * CHN;     // 256
constexpr float THEA    = 0.5f;
constexpr float DELTA   = 1.5f;
constexpr float MINPIX  = 20.0f;

// workspace layout (float offsets)
constexpr int OFF_COUNTS  = 0;                       // KCLS
constexpr int OFF_SUMS    = 32;                      // KCLS*CHN
constexpr int OFF_VAR     = OFF_SUMS + KCLS * CHN;   // KCLS
constexpr int OFF_CENTERS = OFF_VAR + 32;            // KCLS*CHN
constexpr int WS_FLOATS   = OFF_CENTERS + KCLS * CHN;
}

// ---------------- workspace zeroing (graph-replay safe) ----------------
__global__ void k_zero(float* ws, int n) {
  int i = blockIdx.x * 256 + threadIdx.x;
  if (i < n) ws[i] = 0.0f;
}

// ---------------- pixel counts per class ----------------
__global__ __launch_bounds__(256) void k_count(const int* __restrict__ lab,
                                               float* __restrict__ counts) {
  __shared__ unsigned h[KCLS];
  int t = threadIdx.x;
  if (t < KCLS) h[t] = 0u;
  __syncthreads();
  int base = blockIdx.x * 256 * 8;
  #pragma unroll
  for (int i = 0; i < 8; ++i) {
    int p = base + i * 256 + t;
    if (p < NPIX) {
      int l = lab[p];
      if ((unsigned)l < (unsigned)KCLS) atomicAdd(&h[l], 1u);
    }
  }
  __syncthreads();
  if (t < KCLS && h[t]) atomicAdd(&counts[t], (float)h[t]);
}

// ---------------- per-class feature sums (segment sum) ----------------
// grid: (HW/SPAN, PLANES). Per-thread private LDS histogram, stride 21
// (odd -> each lane of a wave32 hits a distinct LDS bank for a fixed class).
// predict is streamed non-temporally: it is single-use per pass and larger
// than the 192MB L2, while the 4MB label array must stay L2-resident since
// it is re-read for each of the 64 channel planes.
__global__ __launch_bounds__(256) void k_sums(const float* __restrict__ x,
                                              const int* __restrict__ lab,
                                              float* __restrict__ sums) {
  constexpr int SPAN = HW / 8;   // 32768 pixels per block
  __shared__ float acc[256 * 21];
  int t = threadIdx.x;
  #pragma unroll
  for (int k = 0; k < 21; ++k) acc[t * 21 + k] = 0.0f;

  int q  = blockIdx.y;           // plane: q = n*CHN + ch
  int n  = q >> 6;
  int ch = q & 63;
  const float* plane = x + (size_t)q * HW;
  const int*   labn  = lab + (size_t)n * HW;
  int base = blockIdx.x * SPAN;

  for (int p = base + t; p < base + SPAN; p += 256) {
    int   l = labn[p];                                   // L2-resident
    float v = __builtin_nontemporal_load(&plane[p]);     // HBM stream, TH=NT
    if ((unsigned)l < (unsigned)KCLS) acc[t * 21 + l] += v;
  }
  __syncthreads();

  for (int off = 128; off > 0; off >>= 1) {
    if (t < off) {
      #pragma unroll
      for (int k = 0; k < KCLS; ++k) acc[t * 21 + k] += acc[(t + off) * 21 + k];
    }
    __syncthreads();
  }
  if (t < KCLS) atomicAdd(&sums[t * CHN + ch], acc[t]);
}

// ---------------- centers = sums / max(counts,1) ----------------
__global__ void k_centers(const float* __restrict__ sums,
                          const float* __restrict__ counts,
                          float* __restrict__ centers) {
  int i = blockIdx.x * 256 + threadIdx.x;
  if (i < KCLS * CHN) {
    float c = counts[i >> 6];
    centers[i] = sums[i] / fmaxf(c, 1.0f);
  }
}

// ---------------- pull (variance) term ----------------
// Thread-per-pixel; centers staged in LDS with row stride 65 so that
// label-indexed gathers across a wave hit distinct banks.
__global__ __launch_bounds__(256) void k_var(const float* __restrict__ x,
                                             const int* __restrict__ lab,
                                             const float* __restrict__ centers,
                                             float* __restrict__ varsum) {
  __shared__ float cpad[KCLS * 65];
  __shared__ float lv[KCLS];
  int t = threadIdx.x;
  for (int i = t; i < KCLS * CHN; i += 256)
    cpad[(i >> 6) * 65 + (i & 63)] = centers[i];
  if (t < KCLS) lv[t] = 0.0f;
  __syncthreads();

  #pragma unroll
  for (int it = 0; it < 2; ++it) {
    int p  = blockIdx.x * 512 + it * 256 + t;
    int n  = p >> 18;                 // HW = 1<<18
    int hw = p & (HW - 1);
    int l  = lab[p];
    if ((unsigned)l < (unsigned)KCLS) {
      const float* px = x + (size_t)(n * CHN) * HW + hw;
      const float* cr = &cpad[l * 65];
      float d2 = 0.0f;
      #pragma unroll
      for (int ch = 0; ch < CHN; ++ch) {
        float xv   = __builtin_nontemporal_load(&px[(size_t)ch * HW]); // NT stream
        float diff = cr[ch] - xv;
        d2 = fmaf(diff, diff, d2);
      }
      float d = sqrtf(fmaxf(d2, 1e-12f));
      float r = fmaxf(d - THEA, 0.0f);
      atomicAdd(&lv[l], r * r);
    }
  }
  __syncthreads();
  if (t < KCLS) atomicAdd(&varsum[t], lv[t]);
}

// ---------------- finalize: Gram via f32 WMMA + scalar losses ----------------
// Launched as a single wave32; EXEC is all-ones for the WMMA ops.
__global__ __launch_bounds__(32) void k_final(const float* __restrict__ counts,
                                              const float* __restrict__ centers,
                                              const float* __restrict__ varsum,
                                              float* __restrict__ out) {
  __shared__ float CS[32][65];   // centers, padded to 32 classes, 0-filled
  __shared__ float G[32][33];    // Gram = CS * CS^T
  __shared__ float vld[32];

  int lane = threadIdx.x;        // 0..31, one wave
  {
    int r = lane;
    for (int col = 0; col < 65; ++col) CS[r][col] = 0.0f;
    if (r < KCLS)
      for (int col = 0; col < CHN; ++col) CS[r][col] = centers[r * CHN + col];
    vld[r] = (r < KCLS && counts[r] > MINPIX) ? 1.0f : 0.0f;
  }
  __syncthreads();

  int mr   = lane & 15;
  int half = lane >> 4;

  // 2x2 tiles of 16x16, K-loop over 64 channels in steps of 4.
  #pragma unroll
  for (int ti = 0; ti < 2; ++ti) {
    #pragma unroll
    for (int tj = 0; tj < 2; ++tj) {
      v8f acc = {};
      #pragma unroll
      for (int kc = 0; kc < CHN; kc += 4) {
        v2f a, b;
        // A 16x4 f32 layout: VGPR0 = K(kc+2*half), VGPR1 = K(kc+2*half+1), M = lane&15
        a.x = CS[ti * 16 + mr][kc + 2 * half];
        a.y = CS[ti * 16 + mr][kc + 2 * half + 1];
        // B 4x16 f32 layout mirrors A with N = lane&15 (B = centers^T)
        b.x = CS[tj * 16 + mr][kc + 2 * half];
        b.y = CS[tj * 16 + mr][kc + 2 * half + 1];
        acc = __builtin_amdgcn_wmma_f32_16x16x4_f32(
            false, a, false, b, (short)0, acc, false, false);
      }
      // C/D layout: VGPR v -> M = v + 8*half, N = lane&15
      #pragma unroll
      for (int v = 0; v < 8; ++v)
        G[ti * 16 + v + 8 * half][tj * 16 + mr] = acc[v];
    }
  }
  __syncthreads();

  // pull-term total (redundant per lane; trivial)
  float n_valid = 0.0f;
  for (int k = 0; k < KCLS; ++k) n_valid += vld[k];
  float loss_var = 0.0f;
  for (int k = 0; k < KCLS; ++k)
    if (vld[k] > 0.0f) loss_var += varsum[k] / fmaxf(counts[k], 1.0f);
  loss_var /= fmaxf(n_valid, 1.0f);

  // push + reg terms from Gram
  float dsum = 0.0f, regsum = 0.0f;
  for (int idx = lane; idx < KCLS * KCLS; idx += 32) {
    int i = idx / KCLS, j = idx % KCLS;
    if (i != j && vld[i] > 0.0f && vld[j] > 0.0f) {
      float sq = G[i][i] + G[j][j] - 2.0f * G[i][j];
      float pd = sqrtf(fmaxf(sq, 0.0f));
      float r  = fmaxf(2.0f * DELTA - pd, 0.0f);
      dsum += r * r;
    }
  }
  for (int k = lane; k < KCLS; k += 32)
    if (vld[k] > 0.0f) regsum += sqrtf(fmaxf(G[k][k], 0.0f));

  for (int o = 16; o > 0; o >>= 1) {
    dsum   += __shfl_xor(dsum, o, 32);
    regsum += __shfl_xor(regsum, o, 32);
  }
  if (lane == 0) {
    float n_pairs = fmaxf(n_valid * (n_valid - 1.0f), 1.0f);
    out[0] = loss_var + dsum / n_pairs + 0.001f * (regsum / fmaxf(n_valid, 1.0f));
  }
}

extern "C" void kernel_launch(void* const* d_in, const int* in_sizes, int n_in,
                              void* d_out, int out_size, void* d_ws, size_t ws_size,
                              hipStream_t stream) {
  (void)in_sizes; (void)n_in; (void)out_size; (void)ws_size;
  const float* predict = (const float*)d_in[0];
  const int*   target  = (const int*)d_in[1];
  float* out = (float*)d_out;
  float* ws  = (float*)d_ws;

  float* counts  = ws + OFF_COUNTS;
  float* sums    = ws + OFF_SUMS;
  float* varsum  = ws + OFF_VAR;
  float* centers = ws + OFF_CENTERS;

  k_zero<<<(WS_FLOATS + 255) / 256, 256, 0, stream>>>(ws, WS_FLOATS);
  k_count<<<NPIX / (256 * 8), 256, 0, stream>>>(target, counts);
  k_sums<<<dim3(8, PLANES), 256, 0, stream>>>(predict, target, sums);
  k_centers<<<(KCLS * CHN + 255) / 256, 256, 0, stream>>>(sums, counts, centers);
  k_var<<<NPIX / 512, 256, 0, stream>>>(predict, target, centers, varsum);
  k_final<<<1, 32, 0, stream>>>(counts, centers, varsum, out);
}